// VSDC_5171140624590
// MI455X (gfx1250) — compile-verified
//
#include <hip/hip_runtime.h>

// ---------------------------------------------------------------------------
// CS-MRI CG-SENSE solver for MI455X (gfx1250).
// Centered ortho 2D FFT implemented as Y = G * X * G with the symmetric
// modified DFT matrix G[k][n] = exp(-2*pi*i*(k-160)(n-160)/320)/sqrt(320).
// Complex GEMMs run on V_WMMA_F32_16X16X4_F32 (full fp32 matrix pipe).
// Wave tile 16x32 (two N-tiles) to reuse A-fragments across columns:
// 8 WMMAs per 6 LDS frag loads per k-step.
// ---------------------------------------------------------------------------

typedef float v2f __attribute__((ext_vector_type(2)));
typedef float v8f __attribute__((ext_vector_type(8)));

#define NS     320          // H == W == 320
#define NS2    (NS * NS)
#define NB     8            // batch
#define NC     16           // coils
#define CCHUNK 4            // coils per chunk (workspace cap)
#define NMAT   (NB * CCHUNK)
#define TM     64           // macro tile rows
#define TN     64           // macro tile cols
#define TK     32           // K slab
#define LDP    (TK + 2)     // LDS row stride: b64-aligned frag loads,
                            // 34*4B stride -> conflict-free frag reads

__device__ __forceinline__ v8f wmma4(v2f a, v2f b, v8f c) {
  // D = A(16x4) * B(4x16) + C(16x16), fp32 in/out.
  return __builtin_amdgcn_wmma_f32_16x16x4_f32(
      /*neg_a=*/false, a, /*neg_b=*/false, b,
      /*c_mod=*/(short)0, c, /*reuse_a=*/false, /*reuse_b=*/false);
}

// ---------------------------------------------------------------------------
// Build modified DFT matrix (exact integer phase reduction: no range issues).
// ---------------------------------------------------------------------------
__global__ __launch_bounds__(256) void k_build_dft(float* __restrict__ Gr,
                                                   float* __restrict__ Gi) {
  int i = blockIdx.x * 256 + threadIdx.x;
  if (i >= NS2) return;
  int k = i / NS, n = i % NS;
  int t = ((k - 160) * (n - 160)) % NS;      // exact, |t| < 320
  float theta = (-6.2831853071795864769f / (float)NS) * (float)t;
  float s, c;
  __sincosf(theta, &s, &c);
  const float scale = 0.05590169943749474241f;  // 1/sqrt(320)
  Gr[i] = c * scale;
  Gi[i] = s * scale;
}

// ---------------------------------------------------------------------------
// r = p = zero_filled + mu*reconstruction  (planar complex);  x (=d_out) = 0
// ---------------------------------------------------------------------------
__global__ __launch_bounds__(256) void k_init(
    const float* __restrict__ zf, const float* __restrict__ rec,
    const float* __restrict__ mu_p,
    float* __restrict__ r_re, float* __restrict__ r_im,
    float* __restrict__ p_re, float* __restrict__ p_im,
    float* __restrict__ xout) {
  int i = blockIdx.x * 256 + threadIdx.x;
  if (i >= NB * NS2) return;
  int b = i / NS2, pix = i % NS2;
  float mu = mu_p[0];
  size_t o = (size_t)b * 2 * NS2 + pix;
  float zr = zf[o] + mu * rec[o];
  float zi = zf[o + NS2] + mu * rec[o + NS2];
  r_re[i] = zr; r_im[i] = zi;
  p_re[i] = zr; p_im[i] = zi;
  xout[o] = 0.0f;
  xout[o + NS2] = 0.0f;
}

// ---------------------------------------------------------------------------
// Per-batch inner product Re<a,b> = sum(ar*br + ai*bi). One block per batch.
// ---------------------------------------------------------------------------
__global__ __launch_bounds__(256) void k_dot(
    const float* __restrict__ ar, const float* __restrict__ ai,
    const float* __restrict__ br, const float* __restrict__ bi,
    float* __restrict__ out) {
  __shared__ float sm[256];
  int b = blockIdx.x;
  const float* Ar = ar + (size_t)b * NS2;
  const float* Ai = ai + (size_t)b * NS2;
  const float* Br = br + (size_t)b * NS2;
  const float* Bi = bi + (size_t)b * NS2;
  float acc = 0.0f;
  for (int i = threadIdx.x; i < NS2; i += 256)
    acc += Ar[i] * Br[i] + Ai[i] * Bi[i];
  sm[threadIdx.x] = acc;
  __syncthreads();
  for (int s = 128; s > 0; s >>= 1) {
    if ((int)threadIdx.x < s) sm[threadIdx.x] += sm[threadIdx.x + s];
    __syncthreads();
  }
  if (threadIdx.x == 0) out[b] = sm[0];
}

// ---------------------------------------------------------------------------
// T[b,cc] = csm[b,c0+cc] * p[b]   (complex elementwise)
// ---------------------------------------------------------------------------
__global__ __launch_bounds__(256) void k_coil_expand(
    const float* __restrict__ p_re, const float* __restrict__ p_im,
    const float* __restrict__ csm, int c0,
    float* __restrict__ tr, float* __restrict__ ti) {
  int i = blockIdx.x * 256 + threadIdx.x;
  if (i >= NMAT * NS2) return;
  int pix = i % NS2;
  int mat = i / NS2;
  int b = mat / CCHUNK;
  int c = c0 + (mat % CCHUNK);
  size_t si = (((size_t)b * NC + c) * NS2 + pix) * 2;  // csm: (B,C,H,W,2)
  float sr = csm[si], sj = csm[si + 1];
  size_t pIdx = (size_t)b * NS2 + pix;
  float pr = p_re[pIdx], pi = p_im[pIdx];
  tr[i] = sr * pr - sj * pi;
  ti[i] = sr * pi + sj * pr;
}

// ---------------------------------------------------------------------------
// Ap[b] (init with mu*p on first chunk) += sum_cc conj(csm) * T
// ---------------------------------------------------------------------------
__global__ __launch_bounds__(256) void k_coil_reduce(
    const float* __restrict__ tr, const float* __restrict__ ti,
    const float* __restrict__ csm, int c0, int init,
    const float* __restrict__ p_re, const float* __restrict__ p_im,
    const float* __restrict__ mu_p,
    float* __restrict__ Ap_re, float* __restrict__ Ap_im) {
  int i = blockIdx.x * 256 + threadIdx.x;
  if (i >= NB * NS2) return;
  int b = i / NS2, pix = i % NS2;
  float ar, ai;
  if (init) {
    float mu = mu_p[0];
    ar = mu * p_re[i];
    ai = mu * p_im[i];
  } else {
    ar = Ap_re[i];
    ai = Ap_im[i];
  }
#pragma unroll
  for (int cc = 0; cc < CCHUNK; ++cc) {
    int c = c0 + cc;
    size_t si = (((size_t)b * NC + c) * NS2 + pix) * 2;
    float sr = csm[si], sj = csm[si + 1];
    size_t t = ((size_t)(b * CCHUNK + cc)) * NS2 + pix;
    float trv = tr[t], tiv = ti[t];
    ar += sr * trv + sj * tiv;   // conj(s)*t
    ai += sr * tiv - sj * trv;
  }
  Ap_re[i] = ar;
  Ap_im[i] = ai;
}

// ---------------------------------------------------------------------------
// CG scalar updates (scalars live on-device -> graph capturable)
// ---------------------------------------------------------------------------
__global__ __launch_bounds__(256) void k_update1(
    float* __restrict__ xout,
    const float* __restrict__ p_re, const float* __restrict__ p_im,
    float* __restrict__ r_re, float* __restrict__ r_im,
    const float* __restrict__ Ap_re, const float* __restrict__ Ap_im,
    const float* __restrict__ s_rr, const float* __restrict__ s_pAp) {
  int i = blockIdx.x * 256 + threadIdx.x;
  if (i >= NB * NS2) return;
  int b = i / NS2, pix = i % NS2;
  float alpha = s_rr[b] / s_pAp[b];
  size_t xo = (size_t)b * 2 * NS2 + pix;
  xout[xo]       += alpha * p_re[i];
  xout[xo + NS2] += alpha * p_im[i];
  r_re[i] -= alpha * Ap_re[i];
  r_im[i] -= alpha * Ap_im[i];
}

__global__ __launch_bounds__(256) void k_update2(
    float* __restrict__ p_re, float* __restrict__ p_im,
    const float* __restrict__ r_re, const float* __restrict__ r_im,
    const float* __restrict__ s_rr2, const float* __restrict__ s_rr) {
  int i = blockIdx.x * 256 + threadIdx.x;
  if (i >= NB * NS2) return;
  int b = i / NS2;
  float beta = s_rr2[b] / s_rr[b];
  p_re[i] = r_re[i] + beta * p_re[i];
  p_im[i] = r_im[i] + beta * p_im[i];
}

// ---------------------------------------------------------------------------
// Complex GEMM: Y[mat] = G * X[mat].  gsign=-1 => conj(G).
// Block 256 thr = 8 waves, macro tile 64x64, wave tile 16x32 (2 N-tiles):
// A-frags reused across both N-tiles -> 8 WMMAs per 6 LDS frag loads.
// ---------------------------------------------------------------------------
__global__ __launch_bounds__(256) void k_gemm_left(
    const float* __restrict__ Gr, const float* __restrict__ Gi, float gsign,
    const float* __restrict__ Xr, const float* __restrict__ Xi,
    float* __restrict__ Yr, float* __restrict__ Yi) {
  __shared__ float Ar[TM][LDP], Ai[TM][LDP];   // G panel (rows m, cols k)
  __shared__ float Br[TN][LDP], Bi[TN][LDP];   // X panel transposed: [n][k]
  const int tid = threadIdx.x, lane = tid & 31, wv = tid >> 5;
  const int wm = wv >> 1, wn = wv & 1;         // wave tile: rows wm*16, cols wn*32
  const size_t mo = (size_t)blockIdx.z * NS2;
  const float* xr = Xr + mo;
  const float* xi = Xi + mo;
  float* yr = Yr + mo;
  float* yi = Yi + mo;
  const int m0 = blockIdx.x * TM, n0 = blockIdx.y * TN;

  v8f arr0 = {}, aii0 = {}, ari0 = {}, air0 = {};
  v8f arr1 = {}, aii1 = {}, ari1 = {}, air1 = {};

  for (int k0 = 0; k0 < NS; k0 += TK) {
    __syncthreads();
    // Stage A = G[m0+r][k0+kk] (contiguous float2 along k)
    for (int idx = tid; idx < TM * (TK / 2); idx += 256) {
      int r = idx / (TK / 2), kk = (idx % (TK / 2)) * 2;
      size_t g = (size_t)(m0 + r) * NS + (k0 + kk);
      float2 vr = *(const float2*)(Gr + g);
      float2 vi = *(const float2*)(Gi + g);
      *(float2*)&Ar[r][kk] = vr;
      float2 wi; wi.x = gsign * vi.x; wi.y = gsign * vi.y;
      *(float2*)&Ai[r][kk] = wi;
    }
    // Stage B = X[k0+kk][n0+nn], scattered into transposed panel [n][k]
    for (int idx = tid; idx < TK * (TN / 2); idx += 256) {
      int kk = idx / (TN / 2), nn = (idx % (TN / 2)) * 2;
      size_t g = (size_t)(k0 + kk) * NS + (n0 + nn);
      float2 vr = *(const float2*)(xr + g);
      float2 vi = *(const float2*)(xi + g);
      Br[nn][kk] = vr.x; Br[nn + 1][kk] = vr.y;
      Bi[nn][kk] = vi.x; Bi[nn + 1][kk] = vi.y;
    }
    __syncthreads();
    const int ml  = wm * 16 + (lane & 15);
    const int nl0 = wn * 32 + (lane & 15);
    const int khalf = (lane >> 4) << 1;   // lanes 16..31 take K+2 (ISA layout)
#pragma unroll
    for (int kk = 0; kk < TK; kk += 4) {
      int klo = kk + khalf;
      v2f a_r  = *(const v2f*)&Ar[ml][klo];
      v2f a_i  = *(const v2f*)&Ai[ml][klo];
      v2f b_r0 = *(const v2f*)&Br[nl0][klo];
      v2f b_i0 = *(const v2f*)&Bi[nl0][klo];
      v2f b_r1 = *(const v2f*)&Br[nl0 + 16][klo];
      v2f b_i1 = *(const v2f*)&Bi[nl0 + 16][klo];
      arr0 = wmma4(a_r, b_r0, arr0);
      aii0 = wmma4(a_i, b_i0, aii0);
      ari0 = wmma4(a_r, b_i0, ari0);
      air0 = wmma4(a_i, b_r0, air0);
      arr1 = wmma4(a_r, b_r1, arr1);
      aii1 = wmma4(a_i, b_i1, aii1);
      ari1 = wmma4(a_r, b_i1, ari1);
      air1 = wmma4(a_i, b_r1, air1);
    }
  }
  // D layout: VGPR v -> row v (lanes 0-15) / v+8 (lanes 16-31), col = lane%16
  const int mg = m0 + wm * 16 + ((lane >> 4) << 3);
  const int ng = n0 + wn * 32 + (lane & 15);
#pragma unroll
  for (int v = 0; v < 8; ++v) {
    size_t o = (size_t)(mg + v) * NS + ng;
    yr[o] = arr0[v] - aii0[v];
    yi[o] = ari0[v] + air0[v];
    yr[o + 16] = arr1[v] - aii1[v];
    yi[o + 16] = ari1[v] + air1[v];
  }
}

// ---------------------------------------------------------------------------
// Complex GEMM: Y[mat] = X[mat] * G (G symmetric -> B panel loads G rows).
// Optionally multiplies the k-space mask in the epilogue (forward pass).
// ---------------------------------------------------------------------------
__global__ __launch_bounds__(256) void k_gemm_right(
    const float* __restrict__ Gr, const float* __restrict__ Gi, float gsign,
    const float* __restrict__ Xr, const float* __restrict__ Xi,
    float* __restrict__ Yr, float* __restrict__ Yi,
    const float* __restrict__ mask, int apply_mask) {
  __shared__ float Ar[TM][LDP], Ai[TM][LDP];   // X rows
  __shared__ float Br[TN][LDP], Bi[TN][LDP];   // G^T panel == G rows (symmetry)
  const int tid = threadIdx.x, lane = tid & 31, wv = tid >> 5;
  const int wm = wv >> 1, wn = wv & 1;
  const int mat = blockIdx.z;
  const size_t mo = (size_t)mat * NS2;
  const float* xr = Xr + mo;
  const float* xi = Xi + mo;
  float* yr = Yr + mo;
  float* yi = Yi + mo;
  const int m0 = blockIdx.x * TM, n0 = blockIdx.y * TN;

  v8f arr0 = {}, aii0 = {}, ari0 = {}, air0 = {};
  v8f arr1 = {}, aii1 = {}, ari1 = {}, air1 = {};

  for (int k0 = 0; k0 < NS; k0 += TK) {
    __syncthreads();
    // Stage A = X[m0+r][k0+kk]
    for (int idx = tid; idx < TM * (TK / 2); idx += 256) {
      int r = idx / (TK / 2), kk = (idx % (TK / 2)) * 2;
      size_t g = (size_t)(m0 + r) * NS + (k0 + kk);
      *(float2*)&Ar[r][kk] = *(const float2*)(xr + g);
      *(float2*)&Ai[r][kk] = *(const float2*)(xi + g);
    }
    // Stage B^T[n][k] = G[k][n] = G[n][k]: direct contiguous loads of G rows
    for (int idx = tid; idx < TN * (TK / 2); idx += 256) {
      int nn = idx / (TK / 2), kk = (idx % (TK / 2)) * 2;
      size_t g = (size_t)(n0 + nn) * NS + (k0 + kk);
      float2 vr = *(const float2*)(Gr + g);
      float2 vi = *(const float2*)(Gi + g);
      *(float2*)&Br[nn][kk] = vr;
      float2 wi; wi.x = gsign * vi.x; wi.y = gsign * vi.y;
      *(float2*)&Bi[nn][kk] = wi;
    }
    __syncthreads();
    const int ml  = wm * 16 + (lane & 15);
    const int nl0 = wn * 32 + (lane & 15);
    const int khalf = (lane >> 4) << 1;
#pragma unroll
    for (int kk = 0; kk < TK; kk += 4) {
      int klo = kk + khalf;
      v2f a_r  = *(const v2f*)&Ar[ml][klo];
      v2f a_i  = *(const v2f*)&Ai[ml][klo];
      v2f b_r0 = *(const v2f*)&Br[nl0][klo];
      v2f b_i0 = *(const v2f*)&Bi[nl0][klo];
      v2f b_r1 = *(const v2f*)&Br[nl0 + 16][klo];
      v2f b_i1 = *(const v2f*)&Bi[nl0 + 16][klo];
      arr0 = wmma4(a_r, b_r0, arr0);
      aii0 = wmma4(a_i, b_i0, aii0);
      ari0 = wmma4(a_r, b_i0, ari0);
      air0 = wmma4(a_i, b_r0, air0);
      arr1 = wmma4(a_r, b_r1, arr1);
      aii1 = wmma4(a_i, b_i1, aii1);
      ari1 = wmma4(a_r, b_i1, ari1);
      air1 = wmma4(a_i, b_r1, air1);
    }
  }
  const int mg = m0 + wm * 16 + ((lane >> 4) << 3);
  const int ng = n0 + wn * 32 + (lane & 15);
  const float* mb = mask + (size_t)(mat / CCHUNK) * NS2;
#pragma unroll
  for (int v = 0; v < 8; ++v) {
    size_t o = (size_t)(mg + v) * NS + ng;
    float mv0 = apply_mask ? mb[o] : 1.0f;
    float mv1 = apply_mask ? mb[o + 16] : 1.0f;
    yr[o] = (arr0[v] - aii0[v]) * mv0;
    yi[o] = (ari0[v] + air0[v]) * mv0;
    yr[o + 16] = (arr1[v] - aii1[v]) * mv1;
    yi[o + 16] = (ari1[v] + air1[v]) * mv1;
  }
}

// ---------------------------------------------------------------------------
// Host: fixed, graph-capturable launch sequence (all scalars stay on device).
// ---------------------------------------------------------------------------
extern "C" void kernel_launch(void* const* d_in, const int* in_sizes, int n_in,
                              void* d_out, int out_size, void* d_ws,
                              size_t ws_size, hipStream_t stream) {
  (void)in_sizes; (void)n_in; (void)out_size; (void)ws_size;
  const float* zf   = (const float*)d_in[0];   // (B,2,H,W)
  const float* rec  = (const float*)d_in[1];   // (B,2,H,W)
  const float* mask = (const float*)d_in[2];   // (B,H,W)
  const float* csm  = (const float*)d_in[3];   // (B,C,H,W,2)
  const float* mu   = (const float*)d_in[4];   // scalar
  float* xout = (float*)d_out;                 // (B,2,H,W); x lives here

  // Workspace layout (floats): ~73 MB total
  float* w = (float*)d_ws;
  float* Gr = w;  w += NS2;
  float* Gi = w;  w += NS2;
  const size_t F = (size_t)NB * NS2;
  float* r_re  = w; w += F;  float* r_im  = w; w += F;
  float* p_re  = w; w += F;  float* p_im  = w; w += F;
  float* Ap_re = w; w += F;  float* Ap_im = w; w += F;
  const size_t CF = (size_t)NMAT * NS2;
  float* bAr = w; w += CF;  float* bAi = w; w += CF;
  float* bBr = w; w += CF;  float* bBi = w; w += CF;
  float* s_rr  = w; w += NB;
  float* s_pAp = w; w += NB;
  float* s_rr2 = w; w += NB;

  const int gF  = (int)((F + 255) / 256);     // 3200 blocks
  const int gCF = (int)((CF + 255) / 256);    // 12800 blocks
  const dim3 gGemm(NS / TM, NS / TN, NMAT);   // (5, 5, 32)

  k_build_dft<<<(NS2 + 255) / 256, 256, 0, stream>>>(Gr, Gi);
  k_init<<<gF, 256, 0, stream>>>(zf, rec, mu, r_re, r_im, p_re, p_im, xout);

  for (int it = 0; it < 10; ++it) {
    k_dot<<<NB, 256, 0, stream>>>(r_re, r_im, r_re, r_im, s_rr);
    for (int c0 = 0; c0 < NC; c0 += CCHUNK) {
      k_coil_expand<<<gCF, 256, 0, stream>>>(p_re, p_im, csm, c0, bAr, bAi);
      // forward: K = G * T * G, then mask
      k_gemm_left <<<gGemm, 256, 0, stream>>>(Gr, Gi,  1.0f, bAr, bAi, bBr, bBi);
      k_gemm_right<<<gGemm, 256, 0, stream>>>(Gr, Gi,  1.0f, bBr, bBi, bAr, bAi, mask, 1);
      // inverse: T2 = conj(G) * K * conj(G)
      k_gemm_left <<<gGemm, 256, 0, stream>>>(Gr, Gi, -1.0f, bAr, bAi, bBr, bBi);
      k_gemm_right<<<gGemm, 256, 0, stream>>>(Gr, Gi, -1.0f, bBr, bBi, bAr, bAi, mask, 0);
      k_coil_reduce<<<gF, 256, 0, stream>>>(bAr, bAi, csm, c0, (c0 == 0) ? 1 : 0,
                                            p_re, p_im, mu, Ap_re, Ap_im);
    }
    k_dot<<<NB, 256, 0, stream>>>(Ap_re, Ap_im, p_re, p_im, s_pAp);
    k_update1<<<gF, 256, 0, stream>>>(xout, p_re, p_im, r_re, r_im,
                                      Ap_re, Ap_im, s_rr, s_pAp);
    k_dot<<<NB, 256, 0, stream>>>(r_re, r_im, r_re, r_im, s_rr2);
    k_update2<<<gF, 256, 0, stream>>>(p_re, p_im, r_re, r_im, s_rr2, s_rr);
  }
}